// SmoothGradientHessianLoss_32083405701214
// MI455X (gfx1250) — compile-verified
//
#include <hip/hip_runtime.h>
#include <stdint.h>

#define TILE 32
#define HH 512
#define WW 512
#define BB 16
#define CC 3
#define NBLK (BB * (HH / TILE) * (WW / TILE)) /* 16*16*16 = 4096 */

__device__ __forceinline__ int refl(int i) {
  if (i < 0) i = -i;
  if (i >= HH) i = 2 * HH - 2 - i;
  return i;
}

// CDNA5 async global->LDS copy (GV mode), tracked by ASYNCcnt.
__device__ __forceinline__ void async_ld_b32(const float* gptr, const float* lptr) {
  uint32_t loff = (uint32_t)(uintptr_t)lptr; // low 32 bits of generic LDS ptr = LDS byte offset
  asm volatile("global_load_async_to_lds_b32 %0, %1, off"
               :: "v"(loff), "v"(gptr)
               : "memory");
}

__device__ __forceinline__ void wait_async0() {
  asm volatile("s_wait_asynccnt 0x0" ::: "memory");
}

__device__ __forceinline__ float sigm(float x) { return 1.f / (1.f + expf(-x)); }

__device__ __forceinline__ void issue_tile_loads(const float* __restrict__ tp,
                                                 const float* __restrict__ pp,
                                                 int y0, int x0, int tid,
                                                 float (*st)[40], float (*sp)[40]) {
  // target tile + halo 3 (38x38), reflect-indexed
  for (int i = tid; i < 38 * 38; i += 256) {
    int r = i / 38, c = i - r * 38;
    async_ld_b32(tp + refl(y0 + r - 3) * WW + refl(x0 + c - 3), &st[r][c]);
  }
  // pred tile + halo 1 (34x34), reflect-indexed
  for (int i = tid; i < 34 * 34; i += 256) {
    int r = i / 34, c = i - r * 34;
    async_ld_b32(pp + refl(y0 + r - 1) * WW + refl(x0 + c - 1), &sp[r][c]);
  }
}

__global__ __launch_bounds__(256) void sghl_main(const float* __restrict__ pred,
                                                 const float* __restrict__ target,
                                                 float* __restrict__ ws) {
  __shared__ float s_t[2][38][40]; // target tile + halo3, double buffered
  __shared__ float s_p[2][34][40]; // pred tile + halo1, double buffered
  __shared__ float s_st[36][40];   // gauss(target), tile + halo2
  __shared__ float s_ix[34][40];   // Scharr-x of st, tile + halo1 (exact reflect values)
  __shared__ float s_iy[34][40];
  __shared__ float r_n[256], r_d[256];

  const int tid = threadIdx.x;
  const int blk = blockIdx.x;
  const int b   = blk >> 8;
  const int t16 = blk & 255;
  const int y0  = (t16 >> 4) * TILE;
  const int x0  = (t16 & 15) * TILE;

  // prologue: async-load channel 0 into buffer 0
  issue_tile_loads(target + (size_t)(b * CC) * HH * WW,
                   pred   + (size_t)(b * CC) * HH * WW,
                   y0, x0, tid, s_t[0], s_p[0]);
  wait_async0();
  __syncthreads();

  float mask_acc[4] = {0.f, 0.f, 0.f, 0.f};
  float hess_acc[4] = {0.f, 0.f, 0.f, 0.f};

  for (int ch = 0; ch < CC; ++ch) {
    const int cur = ch & 1;

    // overlap: kick off next channel's HBM->LDS DMA before computing this one
    if (ch + 1 < CC) {
      issue_tile_loads(target + (size_t)(b * CC + ch + 1) * HH * WW,
                       pred   + (size_t)(b * CC + ch + 1) * HH * WW,
                       y0, x0, tid, s_t[cur ^ 1], s_p[cur ^ 1]);
    }

    // ---- stage A: st = gauss3x3(target), origin -2, 36x36 ----
    {
      const float (*T)[40] = s_t[cur];
      for (int i = tid; i < 36 * 36; i += 256) {
        int r = i / 36, c = i - r * 36;
        float v = T[r][c] + T[r][c + 2] + T[r + 2][c] + T[r + 2][c + 2]
                + 2.f * (T[r][c + 1] + T[r + 1][c] + T[r + 1][c + 2] + T[r + 2][c + 1])
                + 4.f * T[r + 1][c + 1];
        s_st[r][c] = v * 0.0625f;
      }
    }
    __syncthreads();

    // ---- stage B: ix/iy = Scharr(st), origin -1, 34x34; reflect-mapped for exactness ----
    for (int i = tid; i < 34 * 34; i += 256) {
      int r = i / 34, c = i - r * 34;
      int gy = refl(y0 + r - 1), gx = refl(x0 + c - 1);
      int sr = gy - y0 + 2, sc = gx - x0 + 2; // center index in s_st (origin -2)
      float a00 = s_st[sr - 1][sc - 1], a01 = s_st[sr - 1][sc], a02 = s_st[sr - 1][sc + 1];
      float a10 = s_st[sr    ][sc - 1],                          a12 = s_st[sr    ][sc + 1];
      float a20 = s_st[sr + 1][sc - 1], a21 = s_st[sr + 1][sc], a22 = s_st[sr + 1][sc + 1];
      s_ix[r][c] = (3.f * (a02 - a00) + 10.f * (a12 - a10) + 3.f * (a22 - a20)) * 0.03125f;
      s_iy[r][c] = (3.f * (a20 - a00) + 10.f * (a21 - a01) + 3.f * (a22 - a02)) * 0.03125f;
    }
    __syncthreads();

    // ---- stage C: structure tensor + mask + Hessian per output pixel ----
    {
      const float (*P)[40] = s_p[cur];
      const float kG[3][3] = {{1.f, 2.f, 1.f}, {2.f, 4.f, 2.f}, {1.f, 2.f, 1.f}};
#pragma unroll
      for (int p = 0; p < 4; ++p) {
        int idx = p * 256 + tid;
        int oy = idx >> 5, ox = idx & 31;
        float sxx = 0.f, syy = 0.f, sxy = 0.f;
#pragma unroll
        for (int j = 0; j < 3; ++j)
#pragma unroll
          for (int i2 = 0; i2 < 3; ++i2) {
            float w  = kG[j][i2];
            float vx = s_ix[oy + j][ox + i2];
            float vy = s_iy[oy + j][ox + i2];
            sxx += w * vx * vx;
            syy += w * vy * vy;
            sxy += w * vx * vy;
          }
        sxx *= 0.0625f; syy *= 0.0625f; sxy *= 0.0625f;
        float tr  = sxx + syy;
        float dsc = sqrtf(fmaxf(tr * tr - 4.f * (sxx * syy - sxy * sxy), 0.f) + 1e-6f);
        float coh = dsc / (tr + 1e-6f);
        coh *= coh;
        float m = sigm(100.f * (tr - 0.001f)) * sigm(100.f * (0.05f - tr)) * coh;

        float pxx = P[oy + 1][ox] - 2.f * P[oy + 1][ox + 1] + P[oy + 1][ox + 2];
        float pyy = P[oy][ox + 1] - 2.f * P[oy + 1][ox + 1] + P[oy + 2][ox + 1];
        float pxy = 0.25f * (P[oy][ox] - P[oy][ox + 2] - P[oy + 2][ox] + P[oy + 2][ox + 2]);
        float h = sqrtf(pxx * pxx + 2.f * pxy * pxy + pyy * pyy + 1e-6f);

        mask_acc[p] += m;
        hess_acc[p] += h;
      }
    }
    __syncthreads(); // everyone done reading s_p[cur]/s_ix/s_iy before reuse

    if (ch + 1 < CC) {
      wait_async0();   // next channel's tiles fully landed in LDS
      __syncthreads();
    }
  }

  // per-thread partials -> block partials (fixed-order, deterministic)
  float num = 0.f, den = 0.f;
#pragma unroll
  for (int p = 0; p < 4; ++p) {
    float m = mask_acc[p] * (1.f / 3.f); // channel mean
    num += m * hess_acc[p];              // mask * sum_c(hess)
    den += m;
  }
  r_n[tid] = num;
  r_d[tid] = den;
  __syncthreads();
  for (int s = 128; s > 0; s >>= 1) {
    if (tid < s) { r_n[tid] += r_n[tid + s]; r_d[tid] += r_d[tid + s]; }
    __syncthreads();
  }
  if (tid == 0) {
    ws[blk]        = r_n[0];
    ws[NBLK + blk] = r_d[0];
  }
}

__global__ __launch_bounds__(256) void sghl_finalize(const float* __restrict__ ws,
                                                     float* __restrict__ out) {
  __shared__ float sn[256], sd[256];
  int t = threadIdx.x;
  float n = 0.f, d = 0.f;
  for (int i = t; i < NBLK; i += 256) { n += ws[i]; d += ws[NBLK + i]; }
  sn[t] = n; sd[t] = d;
  __syncthreads();
  for (int s = 128; s > 0; s >>= 1) {
    if (t < s) { sn[t] += sn[t + s]; sd[t] += sd[t + s]; }
    __syncthreads();
  }
  if (t == 0) out[0] = sn[0] / fmaxf(sd[0], 1e-6f);
}

extern "C" void kernel_launch(void* const* d_in, const int* in_sizes, int n_in,
                              void* d_out, int out_size, void* d_ws, size_t ws_size,
                              hipStream_t stream) {
  const float* pred   = (const float*)d_in[0];
  const float* target = (const float*)d_in[1];
  float* out = (float*)d_out;
  float* ws  = (float*)d_ws; // needs 2*NBLK*4 = 32 KB

  sghl_main<<<NBLK, 256, 0, stream>>>(pred, target, ws);
  sghl_finalize<<<1, 256, 0, stream>>>(ws, out);
}